// SPRGNN_88648124990884
// MI455X (gfx1250) — compile-verified
//
#include <hip/hip_runtime.h>

typedef __attribute__((ext_vector_type(2))) float v2f;
typedef __attribute__((ext_vector_type(8))) float v8f;

#define N_NODES   50000
#define N_EDGES   1600000
#define EMB       64
#define HID       128
#define HEADS     4
#define HC        32
#define NCLS      2
#define NGRAPH    512
#define NEG_SLOPE 0.2f
#define GAT_EPS   1e-16f
#define E_TOT     (N_EDGES + N_NODES)   // edges + self loops

// ---------------------------------------------------------------------------
// helpers
// ---------------------------------------------------------------------------
__device__ __forceinline__ void atomicMaxFloat(float* addr, float value) {
  // Exact float ordering via int/uint atomics; correct for any current value
  // (including -inf initialization).
  if (value >= 0.0f)
    atomicMax((int*)addr, __float_as_int(value));
  else
    atomicMin((unsigned int*)addr, __float_as_uint(value));
}

__device__ __forceinline__ float edge_logit(int e, int hd,
                                            const int* __restrict__ src,
                                            const int* __restrict__ dst,
                                            const float* __restrict__ as_n,
                                            const float* __restrict__ ad_n,
                                            int& s, int& d) {
  if (e < N_EDGES) { s = src[e]; d = dst[e]; }
  else             { s = e - N_EDGES; d = s; }          // self loop
  float v = as_n[s * HEADS + hd] + ad_n[d * HEADS + hd];
  return v > 0.0f ? v : NEG_SLOPE * v;                  // leaky_relu
}

// ---------------------------------------------------------------------------
// kernels
// ---------------------------------------------------------------------------
__global__ void k_fill(float* __restrict__ p, float v, int n) {
  int i = blockIdx.x * blockDim.x + threadIdx.x;
  if (i < n) p[i] = v;
}

__global__ void k_embed(const float* __restrict__ emb, const int* __restrict__ tok,
                        float* __restrict__ x) {
  int i = blockIdx.x * blockDim.x + threadIdx.x;        // over N*EMB
  if (i >= N_NODES * EMB) return;
  int nd = i / EMB, c = i - nd * EMB;
  x[i] = emb[tok[nd] * EMB + c];                        // padding row 0 is already 0
}

// H[N,128] = X[N,IN] @ W[IN,128] via V_WMMA_F32_16X16X4_F32.
// Block = 256 threads = 8 waves; block computes a 16-row x 128-col tile.
// Wave w owns columns [16w, 16w+16). X tile + full W staged in LDS.
template <int IN>
__global__ __launch_bounds__(256) void k_gemm_wmma(const float* __restrict__ X,
                                                   const float* __restrict__ W,
                                                   float* __restrict__ H) {
  __shared__ float Xs[16 * IN];
  __shared__ float Ws[IN * HID];
  const int t = threadIdx.x;
  const int rowbase = blockIdx.x * 16;

  for (int i = t * 4; i < IN * HID; i += 256 * 4)
    *(float4*)&Ws[i] = *(const float4*)&W[i];
  for (int i = t * 4; i < 16 * IN; i += 256 * 4) {
    int r = i / IN, c = i - r * IN;
    *(float4*)&Xs[i] = *(const float4*)&X[(rowbase + r) * IN + c];
  }
  __syncthreads();

  const int wave    = t >> 5;       // 0..7 -> column tile
  const int lane    = t & 31;
  const int halfsel = lane >> 4;    // 0 or 1
  const int l15     = lane & 15;
  const int col0    = wave * 16;

  v8f acc = {};
  #pragma unroll 4
  for (int k0 = 0; k0 < IN; k0 += 4) {
    const int ka = k0 + 2 * halfsel;
    v2f a, b;
    // A 16x4: lane l -> row l&15, K = k0 + 2*(l>>4) + {0,1}
    a.x = Xs[l15 * IN + ka];
    a.y = Xs[l15 * IN + ka + 1];
    // B 4x16: VGPR0 -> row ka, VGPR1 -> row ka+1, N = l&15
    b.x = Ws[ka * HID + col0 + l15];
    b.y = Ws[(ka + 1) * HID + col0 + l15];
    acc = __builtin_amdgcn_wmma_f32_16x16x4_f32(false, a, false, b,
                                                (short)0, acc, false, false);
  }

  // C/D 16x16 f32: VGPR j -> row j (lanes 0-15) / j+8 (lanes 16-31), N = l&15
  #pragma unroll
  for (int j = 0; j < 8; ++j)
    H[(rowbase + j + 8 * halfsel) * HID + col0 + l15] = acc[j];
}

__global__ void k_alpha(const float* __restrict__ H,
                        const float* __restrict__ a_s, const float* __restrict__ a_d,
                        float* __restrict__ as_n, float* __restrict__ ad_n) {
  int i = blockIdx.x * blockDim.x + threadIdx.x;        // over N*HEADS
  if (i >= N_NODES * HEADS) return;
  int nd = i >> 2, hd = i & 3;
  const float* hp = &H[nd * HID + hd * HC];
  const float* ws = &a_s[hd * HC];
  const float* wd = &a_d[hd * HC];
  float ss = 0.0f, sd = 0.0f;
  #pragma unroll
  for (int c = 0; c < HC; ++c) { float h = hp[c]; ss += h * ws[c]; sd += h * wd[c]; }
  as_n[i] = ss; ad_n[i] = sd;
}

__global__ void k_edge_max(const int* __restrict__ src, const int* __restrict__ dst,
                           const float* __restrict__ as_n, const float* __restrict__ ad_n,
                           float* __restrict__ mmax) {
  int i = blockIdx.x * blockDim.x + threadIdx.x;        // over E_TOT*HEADS
  if (i >= E_TOT * HEADS) return;
  int e = i >> 2, hd = i & 3;
  int s, d;
  float v = edge_logit(e, hd, src, dst, as_n, ad_n, s, d);
  atomicMaxFloat(&mmax[d * HEADS + hd], v);
}

__global__ void k_edge_sum(const int* __restrict__ src, const int* __restrict__ dst,
                           const float* __restrict__ as_n, const float* __restrict__ ad_n,
                           const float* __restrict__ mmax, float* __restrict__ denom) {
  int i = blockIdx.x * blockDim.x + threadIdx.x;
  if (i >= E_TOT * HEADS) return;
  int e = i >> 2, hd = i & 3;
  int s, d;
  float v = edge_logit(e, hd, src, dst, as_n, ad_n, s, d);
  atomicAdd(&denom[d * HEADS + hd], expf(v - mmax[d * HEADS + hd]));
}

__global__ void k_edge_agg(const int* __restrict__ src, const int* __restrict__ dst,
                           const float* __restrict__ as_n, const float* __restrict__ ad_n,
                           const float* __restrict__ mmax, const float* __restrict__ denom,
                           const float* __restrict__ H, float* __restrict__ out) {
  int i = blockIdx.x * blockDim.x + threadIdx.x;        // over E_TOT*HEADS
  if (i >= E_TOT * HEADS) return;
  int e = i >> 2, hd = i & 3;
  int s, d;
  float v = edge_logit(e, hd, src, dst, as_n, ad_n, s, d);
  int base = d * HEADS + hd;
  float alpha = expf(v - mmax[base]) / (denom[base] + GAT_EPS);
  const float4* hp = (const float4*)&H[s * HID + hd * HC];
  float* op = &out[d * HID + hd * HC];
  #pragma unroll
  for (int q = 0; q < HC / 4; ++q) {
    float4 hv = hp[q];
    atomicAdd(&op[4 * q + 0], hv.x * alpha);
    atomicAdd(&op[4 * q + 1], hv.y * alpha);
    atomicAdd(&op[4 * q + 2], hv.z * alpha);
    atomicAdd(&op[4 * q + 3], hv.w * alpha);
  }
}

__global__ void k_bias_relu(float* __restrict__ x, const float* __restrict__ b) {
  int i = blockIdx.x * blockDim.x + threadIdx.x;        // over N*HID
  if (i >= N_NODES * HID) return;
  float v = x[i] + b[i & (HID - 1)];
  x[i] = v > 0.0f ? v : 0.0f;
}

__global__ void k_pool(const float* __restrict__ x, const int* __restrict__ batch,
                       float* __restrict__ pooled, float* __restrict__ cnt) {
  int i = blockIdx.x * blockDim.x + threadIdx.x;        // over N*HID
  if (i >= N_NODES * HID) return;
  int nd = i >> 7, c = i & (HID - 1);
  int g = batch[nd];
  atomicAdd(&pooled[g * HID + c], x[i]);
  if (c == 0) atomicAdd(&cnt[g], 1.0f);
}

__global__ void k_final(const float* __restrict__ pooled, const float* __restrict__ cnt,
                        const float* __restrict__ lw, const float* __restrict__ lb,
                        float* __restrict__ out) {
  int i = blockIdx.x * blockDim.x + threadIdx.x;        // over NGRAPH*NCLS
  if (i >= NGRAPH * NCLS) return;
  int g = i / NCLS, k = i - g * NCLS;
  float cn = cnt[g]; cn = cn > 1.0f ? cn : 1.0f;
  float s = 0.0f;
  #pragma unroll 8
  for (int j = 0; j < HID; ++j) s += (pooled[g * HID + j] / cn) * lw[j * NCLS + k];
  out[i] = s + lb[k];
}

// ---------------------------------------------------------------------------
// launcher
// ---------------------------------------------------------------------------
static inline int cdiv(long long a, int b) { return (int)((a + b - 1) / b); }

extern "C" void kernel_launch(void* const* d_in, const int* in_sizes, int n_in,
                              void* d_out, int out_size, void* d_ws, size_t ws_size,
                              hipStream_t stream) {
  const float* emb_t = (const float*)d_in[0];
  const float* W1    = (const float*)d_in[1];
  const float* as1   = (const float*)d_in[2];
  const float* ad1   = (const float*)d_in[3];
  const float* b1    = (const float*)d_in[4];
  const float* W2    = (const float*)d_in[5];
  const float* as2   = (const float*)d_in[6];
  const float* ad2   = (const float*)d_in[7];
  const float* b2    = (const float*)d_in[8];
  const float* W3    = (const float*)d_in[9];
  const float* as3   = (const float*)d_in[10];
  const float* ad3   = (const float*)d_in[11];
  const float* b3    = (const float*)d_in[12];
  const float* linW  = (const float*)d_in[13];
  const float* linb  = (const float*)d_in[14];
  const int*   tok   = (const int*)d_in[15];
  const int*   src   = (const int*)d_in[16];
  const int*   dst   = (const int*)d_in[17];
  const int*   batch = (const int*)d_in[18];
  float* out = (float*)d_out;

  // workspace carve-up (256B aligned regions)
  char* base = (char*)d_ws;
  size_t off = 0;
  auto carve = [&](size_t bytes) -> float* {
    float* p = (float*)(base + off);
    off += (bytes + 255) & ~(size_t)255;
    return p;
  };
  float* x0     = carve((size_t)N_NODES * EMB * 4);
  float* hbuf   = carve((size_t)N_NODES * HID * 4);
  float* xa     = carve((size_t)N_NODES * HID * 4);
  float* xb     = carve((size_t)N_NODES * HID * 4);
  float* as_n   = carve((size_t)N_NODES * HEADS * 4);
  float* ad_n   = carve((size_t)N_NODES * HEADS * 4);
  float* mmax   = carve((size_t)N_NODES * HEADS * 4);
  float* denom  = carve((size_t)N_NODES * HEADS * 4);
  float* pooled = carve((size_t)NGRAPH * HID * 4);
  float* cnt    = carve((size_t)NGRAPH * 4);
  (void)ws_size; (void)n_in; (void)in_sizes; (void)out_size;

  const int T = 256;
  const int nNH  = N_NODES * HEADS;
  const int nNC  = N_NODES * HID;
  const int nEH  = E_TOT * HEADS;
  const int gemmBlocks = N_NODES / 16;   // 3125, exact

  // embedding gather
  k_embed<<<cdiv((long long)N_NODES * EMB, T), T, 0, stream>>>(emb_t, tok, x0);

  struct LayerArgs {
    const float *W, *as_w, *ad_w, *bias;
    const float* xin; float* xout; int in_dim;
  } layers[3] = {
    { W1, as1, ad1, b1, x0, xa,  EMB },
    { W2, as2, ad2, b2, xa, xb,  HID },
    { W3, as3, ad3, b3, xb, xa,  HID },
  };

  for (int L = 0; L < 3; ++L) {
    const LayerArgs& A = layers[L];
    if (A.in_dim == EMB)
      k_gemm_wmma<EMB><<<gemmBlocks, T, 0, stream>>>(A.xin, A.W, hbuf);
    else
      k_gemm_wmma<HID><<<gemmBlocks, T, 0, stream>>>(A.xin, A.W, hbuf);

    k_alpha<<<cdiv(nNH, T), T, 0, stream>>>(hbuf, A.as_w, A.ad_w, as_n, ad_n);

    k_fill<<<cdiv(nNH, T), T, 0, stream>>>(mmax, -INFINITY, nNH);
    k_fill<<<cdiv(nNH, T), T, 0, stream>>>(denom, 0.0f, nNH);
    k_fill<<<cdiv(nNC, T), T, 0, stream>>>(A.xout, 0.0f, nNC);

    k_edge_max<<<cdiv(nEH, T), T, 0, stream>>>(src, dst, as_n, ad_n, mmax);
    k_edge_sum<<<cdiv(nEH, T), T, 0, stream>>>(src, dst, as_n, ad_n, mmax, denom);
    k_edge_agg<<<cdiv(nEH, T), T, 0, stream>>>(src, dst, as_n, ad_n, mmax, denom,
                                               hbuf, A.xout);
    k_bias_relu<<<cdiv(nNC, T), T, 0, stream>>>(A.xout, A.bias);
  }

  // mean pool + classifier
  k_fill<<<cdiv(NGRAPH * HID, T), T, 0, stream>>>(pooled, 0.0f, NGRAPH * HID);
  k_fill<<<cdiv(NGRAPH, T), T, 0, stream>>>(cnt, 0.0f, NGRAPH);
  k_pool<<<cdiv(nNC, T), T, 0, stream>>>(xa, batch, pooled, cnt);
  k_final<<<cdiv(NGRAPH * NCLS, T), T, 0, stream>>>(pooled, cnt, linW, linb, out);
}